// DGSSMLoss_5411658793136
// MI455X (gfx1250) — compile-verified
//
#include <hip/hip_runtime.h>

typedef float v2f __attribute__((ext_vector_type(2)));
typedef float v8f __attribute__((ext_vector_type(8)));

#define B_ 16
#define N_ 4096
#define M_ 4096
#define K_ 128
#define INF_BITS 0x7F800000u
#define TILES 4          // pred tiles (of 16) per wave
#define WAVES 8          // waves per block
#define PRED_PER_BLOCK (WAVES * TILES * 16)   // 512
#define CHUNK 2048       // targets staged in LDS per pass

// ---------------------------------------------------------------------------
// Kernel 1: apply Euler-XYZ rotation + scale + translation to targ_shape,
// store padded float4 {x, y, z, |t|^2}.
// ---------------------------------------------------------------------------
__global__ void dgssm_transform_targ(const float* __restrict__ targ_shape,
                                     const float* __restrict__ targ_affine,
                                     float4* __restrict__ targq) {
    const int b = blockIdx.x;
    const float* af = targ_affine + b * 9;
    const float ax = af[0], ay = af[1], az = af[2];
    const float tx = af[3], ty = af[4], tz = af[5];
    const float s0 = af[6], s1 = af[7], s2 = af[8];
    const float cx = cosf(ax), sx = sinf(ax);
    const float cy = cosf(ay), sy = sinf(ay);
    const float cz = cosf(az), sz = sinf(az);
    // R = Rx @ Ry @ Rz
    const float R00 = cy * cz,                 R01 = -cy * sz,                R02 = sy;
    const float R10 = cx * sz + sx * sy * cz,  R11 = cx * cz - sx * sy * sz,  R12 = -sx * cy;
    const float R20 = sx * sz - cx * sy * cz,  R21 = sx * cz + cx * sy * sz,  R22 = cx * cy;

    const float* ts = targ_shape + (size_t)b * M_ * 3;
    float4* out = targq + (size_t)b * M_;
    for (int m = threadIdx.x; m < M_; m += blockDim.x) {
        const float x = ts[m * 3 + 0], y = ts[m * 3 + 1], z = ts[m * 3 + 2];
        const float ox = (x * R00 + y * R10 + z * R20) * s0 + tx;
        const float oy = (x * R01 + y * R11 + z * R21) * s1 + ty;
        const float oz = (x * R02 + y * R12 + z * R22) * s2 + tz;
        out[m] = make_float4(ox, oy, oz, ox * ox + oy * oy + oz * oz);
    }
}

// ---------------------------------------------------------------------------
// Kernel 2: init per-target global min array (+inf as uint bits).
// ---------------------------------------------------------------------------
__global__ void dgssm_init_minT(unsigned* __restrict__ minT, int n) {
    const int i = blockIdx.x * blockDim.x + threadIdx.x;
    if (i < n) minT[i] = INF_BITS;
}

// ---------------------------------------------------------------------------
// Kernel 3: Chamfer via V_WMMA_F32_16X16X4_F32.
// 256 threads = 8 wave32; each wave owns 4 register-resident 16x4 A tiles
// (-2*pred folded in). Targets stream through LDS (2 x 2048 chunk passes,
// +16 pad rows for the software-pipelined over-read). Per 16-target tile:
// 1 ds_load_b128 per lane, 4 WMMAs, running pred-min in registers, and one
// unconditional ds_min_u32 per lane for the target side (half-wave partial
// mins are valid since max(.,0) commutes with min; no cross-lane combine).
//
// float_control(precise, off): values are NaN-free by construction, so let
// llvm.minnum lower to bare v_min_num without the sNaN-quieting
// v_max_num x,x canonicalizations, and fold the +inf min-identity seeds.
// ---------------------------------------------------------------------------
#pragma float_control(push)
#pragma float_control(precise, off)
__launch_bounds__(256)
__global__ void dgssm_chamfer_wmma(const float* __restrict__ pred_shape,
                                   const float4* __restrict__ targq,
                                   float* __restrict__ minP,
                                   unsigned* __restrict__ minT) {
    __shared__ float4   Tq[CHUNK + 16];   // 33 KB staged targets (+pad row)
    __shared__ unsigned minTl[M_];        // 16 KB per-block per-target min

    const int b    = blockIdx.y;
    const int tid  = threadIdx.x;
    const int wave = tid >> 5;
    const int lane = tid & 31;
    const int col  = lane & 15;
    const int hi   = lane >> 4;           // 0: K=0,1 / rows j ; 1: K=2,3 / rows j+8
    const bool lo  = (hi == 0);
    const int predBase = blockIdx.x * PRED_PER_BLOCK + wave * (TILES * 16);

    for (int i = tid; i < M_; i += 256) minTl[i] = INF_BITS;

    // --- A operands (TILES 16x4 tiles, -2 folded in) + per-lane row norms ---
    v2f   a[TILES];
    float p2r[TILES][8], acc[TILES][8];
#pragma unroll
    for (int i = 0; i < TILES; ++i) {
        const float* P = pred_shape + ((size_t)b * N_ + predBase + i * 16 + col) * 3;
        const float x = P[0], y = P[1], z = P[2];
        const float s  = lo ? (x * x + y * y) : (z * z);
        const float p2 = s + __shfl_xor(s, 16, 32);
        a[i][0] = lo ? -2.f * x : -2.f * z;
        a[i][1] = lo ? -2.f * y : 0.f;     // K=3 pad row = 0
#pragma unroll
        for (int j = 0; j < 8; ++j) p2r[i][j] = __shfl(p2, j + 8 * hi, 32);
#pragma unroll
        for (int j = 0; j < 8; ++j) acc[i][j] = __builtin_inff();
    }

    const float4* tb = targq + (size_t)b * M_;
    for (int chunk = 0; chunk < M_ / CHUNK; ++chunk) {
        __syncthreads();                              // Tq reuse + minTl init
        for (int i = tid; i < CHUNK; i += 256) Tq[i] = tb[chunk * CHUNK + i];
        __syncthreads();

        float4 T = Tq[col];                           // software pipeline head
        for (int mt = 0; mt < CHUNK / 16; ++mt) {
            const float4 Tn = Tq[(mt + 1) * 16 + col];  // prefetch next tile (pad at end)
            v2f bv;
            bv[0] = lo ? T.x : T.z;
            bv[1] = lo ? T.y : 0.f;
            const float t2 = T.w;

            v8f c = {};
            v8f d[TILES];
#pragma unroll
            for (int i = 0; i < TILES; ++i)
                d[i] = __builtin_amdgcn_wmma_f32_16x16x4_f32(
                    false, a[i], false, bv, (short)0, c, false, false);

            float tp = __builtin_inff();
#pragma unroll
            for (int i = 0; i < TILES; ++i) {
#pragma unroll
                for (int j = 0; j < 8; ++j) {
                    const float e = d[i][j];
                    acc[i][j] = fminf(acc[i][j], e + t2);   // pred-side running min
                    tp = fminf(tp, p2r[i][j] + e);          // targ-side partial (half rows)
                }
            }
            // Both half-waves publish their clamped partial min: one
            // unconditional LDS atomic per lane, no cross-lane combine.
            const float v = fmaxf(tp + t2, 0.f);
            atomicMin(&minTl[chunk * CHUNK + mt * 16 + col], __float_as_uint(v));
            T = Tn;
        }
    }

    // pred-side: reduce across the 16 columns of each half-wave
#pragma unroll
    for (int m = 1; m <= 8; m <<= 1) {
#pragma unroll
        for (int i = 0; i < TILES; ++i)
#pragma unroll
            for (int j = 0; j < 8; ++j)
                acc[i][j] = fminf(acc[i][j], __shfl_xor(acc[i][j], m, 32));
    }
    if (col == 0) {                                   // lanes 0 and 16
        float* outP = minP + (size_t)b * N_ + predBase;
#pragma unroll
        for (int i = 0; i < TILES; ++i)
#pragma unroll
            for (int j = 0; j < 8; ++j)
                outP[i * 16 + j + 8 * hi] = fmaxf(p2r[i][j] + acc[i][j], 0.f);
    }

    __syncthreads();
    unsigned* outT = minT + (size_t)b * M_;
    for (int i = tid; i < M_; i += 256) atomicMin(&outT[i], minTl[i]);
}
#pragma float_control(pop)

// ---------------------------------------------------------------------------
// Kernel 4: final reduction -> (total, point, coeff, affine)
// ---------------------------------------------------------------------------
__launch_bounds__(256)
__global__ void dgssm_finalize(const float* __restrict__ minP,
                               const unsigned* __restrict__ minT,
                               const float* __restrict__ pw,
                               const float* __restrict__ tw,
                               const float* __restrict__ pa,
                               const float* __restrict__ ta,
                               float* __restrict__ out) {
    __shared__ float red[256];
    const int tid = threadIdx.x;
    float sP = 0.f, sT = 0.f, sC = 0.f, sA = 0.f;
    for (int i = tid; i < B_ * N_; i += 256) sP += minP[i];
    for (int i = tid; i < B_ * M_; i += 256) sT += __uint_as_float(minT[i]);
    for (int i = tid; i < B_ * K_; i += 256) { const float d = pw[i] - tw[i]; sC += d * d; }
    for (int i = tid; i < B_ * 9;  i += 256) { const float d = pa[i] - ta[i]; sA += d * d; }

    auto block_sum = [&](float v) -> float {
        red[tid] = v; __syncthreads();
        for (int s = 128; s > 0; s >>= 1) {
            if (tid < s) red[tid] += red[tid + s];
            __syncthreads();
        }
        const float r = red[0]; __syncthreads();
        return r;
    };
    const float rP = block_sum(sP);
    const float rT = block_sum(sT);
    const float rC = block_sum(sC);
    const float rA = block_sum(sA);
    if (tid == 0) {
        const float point = rP / (float)(B_ * N_) + rT / (float)(B_ * M_);
        const float coeff = rC / (float)(B_ * K_);
        const float aff   = rA / (float)(B_ * 9);
        out[0] = point + 0.5f * coeff + 0.5f * aff;
        out[1] = point;
        out[2] = coeff;
        out[3] = aff;
    }
}

// ---------------------------------------------------------------------------
extern "C" void kernel_launch(void* const* d_in, const int* in_sizes, int n_in,
                              void* d_out, int out_size, void* d_ws, size_t ws_size,
                              hipStream_t stream) {
    const float* pred_shape = (const float*)d_in[0];
    const float* pred_w     = (const float*)d_in[1];
    const float* pred_a     = (const float*)d_in[2];
    const float* targ_shape = (const float*)d_in[3];
    const float* targ_w     = (const float*)d_in[4];
    const float* targ_a     = (const float*)d_in[5];

    // ws layout: targq float4[B*M] | minP float[B*N] | minT uint[B*M]
    float4*   targq = (float4*)d_ws;
    float*    minP  = (float*)((char*)d_ws + sizeof(float4) * (size_t)B_ * M_);
    unsigned* minT  = (unsigned*)(minP + (size_t)B_ * N_);

    dgssm_transform_targ<<<B_, 256, 0, stream>>>(targ_shape, targ_a, targq);
    dgssm_init_minT<<<(B_ * M_ + 255) / 256, 256, 0, stream>>>(minT, B_ * M_);
    dgssm_chamfer_wmma<<<dim3(N_ / PRED_PER_BLOCK, B_), 256, 0, stream>>>(
        pred_shape, targq, minP, minT);
    dgssm_finalize<<<1, 256, 0, stream>>>(minP, minT, pred_w, targ_w, pred_a, targ_a,
                                          (float*)d_out);
}